// Structure2Vec_65670049956016
// MI455X (gfx1250) — compile-verified
//
#include <hip/hip_runtime.h>

// Structure2Vec on gfx1250 (MI455X): N=1024 nodes, E=64 embedding.
// WMMA (V_WMMA_F32_16X16X4_F32) used for the h2 = emb @ W7^T GEMM, fused
// with bias+relu+W5-dot into the final q output.

#define N_NODES 1024
#define EDIM    64
#define NUM_UPD 4

typedef __attribute__((ext_vector_type(2))) float v2f;
typedef __attribute__((ext_vector_type(8))) float v8f;

// ---------------------------------------------------------------------------
// Kernel 1: base[j,e] = x[j]*W1[e] + b1[e] + sum_i relu(C[i,j]*w4[e] + b4[e])
// 64 blocks x 256 threads. Block b owns 16 columns j; each thread owns one j
// and 4 consecutive e's (4 register accumulators), streaming all 1024 rows.
// Each C element is read exactly once per block (broadcast across e-threads).
// ---------------------------------------------------------------------------
__global__ __launch_bounds__(256) void base_kernel(
    const float* __restrict__ x,    // nodes_features (1,N)
    const float* __restrict__ C,    // nodes_cost (N,N) row-major
    const float* __restrict__ W1,   // (E,1)
    const float* __restrict__ b1,   // (E)
    const float* __restrict__ W4,   // (E,1)
    const float* __restrict__ b4,   // (E)
    float* __restrict__ base)       // (N,E)
{
    const int tid = threadIdx.x;
    const int j   = blockIdx.x * 16 + (tid >> 4);
    const int e0  = (tid & 15) * 4;

    const float w40 = W4[e0 + 0], w41 = W4[e0 + 1];
    const float w42 = W4[e0 + 2], w43 = W4[e0 + 3];
    const float b40 = b4[e0 + 0], b41 = b4[e0 + 1];
    const float b42 = b4[e0 + 2], b43 = b4[e0 + 3];

    float a0 = 0.f, a1 = 0.f, a2 = 0.f, a3 = 0.f;
#pragma unroll 4
    for (int i = 0; i < N_NODES; ++i) {
        const float c = C[i * N_NODES + j];
        a0 += fmaxf(fmaf(c, w40, b40), 0.f);
        a1 += fmaxf(fmaf(c, w41, b41), 0.f);
        a2 += fmaxf(fmaf(c, w42, b42), 0.f);
        a3 += fmaxf(fmaf(c, w43, b43), 0.f);
    }

    const float xj = x[j];
    float4 r;
    r.x = a0 + fmaf(xj, W1[e0 + 0], b1[e0 + 0]);
    r.y = a1 + fmaf(xj, W1[e0 + 1], b1[e0 + 1]);
    r.z = a2 + fmaf(xj, W1[e0 + 2], b1[e0 + 2]);
    r.w = a3 + fmaf(xj, W1[e0 + 3], b1[e0 + 3]);
    *(float4*)(base + j * EDIM + e0) = r;
}

// ---------------------------------------------------------------------------
// Kernel 2: emb[j,e] = relu(base[j,e] + v2[e]).  (First iteration: v2 == b2.)
// 256 blocks x 256 threads, one element each (coalesced).
// ---------------------------------------------------------------------------
__global__ __launch_bounds__(256) void update_kernel(
    const float* __restrict__ base,
    const float* __restrict__ v2,   // (E)
    float* __restrict__ emb)        // (N,E)
{
    const int idx = blockIdx.x * 256 + threadIdx.x;
    emb[idx] = fmaxf(base[idx] + v2[idx & (EDIM - 1)], 0.f);
}

// ---------------------------------------------------------------------------
// Kernel 3: s[e] = sum_j emb[j,e];  v2[e] = b2[e] + dot(W2[e,:], s)
// Single block, 256 threads: thread (g=tid/64, e=tid%64) sums a 256-row chunk.
// ---------------------------------------------------------------------------
__global__ __launch_bounds__(256) void reduce_v2_kernel(
    const float* __restrict__ emb,
    const float* __restrict__ W2,   // (E,E)
    const float* __restrict__ b2,   // (E)
    float* __restrict__ v2)         // (E)
{
    __shared__ float part[256];
    __shared__ float sld[EDIM];
    const int tid = threadIdx.x;
    const int e = tid & (EDIM - 1);
    const int g = tid >> 6;

    float acc = 0.f;
    for (int j = g * 256; j < (g + 1) * 256; ++j) acc += emb[j * EDIM + e];
    part[tid] = acc;
    __syncthreads();

    if (tid < EDIM)
        sld[tid] = part[tid] + part[tid + 64] + part[tid + 128] + part[tid + 192];
    __syncthreads();

    if (tid < EDIM) {
        float v = b2[tid];
        const float* wr = W2 + tid * EDIM;
#pragma unroll 8
        for (int k = 0; k < EDIM; ++k) v = fmaf(wr[k], sld[k], v);
        v2[tid] = v;
    }
}

// ---------------------------------------------------------------------------
// Kernel 4: qconst = b5 + sum_e W5[0,e] * relu( (W6 @ s + b6)[e] ),
// where s = emb.sum(axis=0).  Single block.
// ---------------------------------------------------------------------------
__global__ __launch_bounds__(256) void final_head_kernel(
    const float* __restrict__ emb,
    const float* __restrict__ W6,   // (E,E)
    const float* __restrict__ b6,   // (E)
    const float* __restrict__ W5,   // (1,2E)
    const float* __restrict__ b5,   // (1)
    float* __restrict__ qconst)     // scalar
{
    __shared__ float part[256];
    __shared__ float sld[EDIM];
    __shared__ float hq[EDIM];
    const int tid = threadIdx.x;
    const int e = tid & (EDIM - 1);
    const int g = tid >> 6;

    float acc = 0.f;
    for (int j = g * 256; j < (g + 1) * 256; ++j) acc += emb[j * EDIM + e];
    part[tid] = acc;
    __syncthreads();

    if (tid < EDIM)
        sld[tid] = part[tid] + part[tid + 64] + part[tid + 128] + part[tid + 192];
    __syncthreads();

    if (tid < EDIM) {
        float v = b6[tid];
        const float* wr = W6 + tid * EDIM;
#pragma unroll 8
        for (int k = 0; k < EDIM; ++k) v = fmaf(wr[k], sld[k], v);
        hq[tid] = W5[tid] * fmaxf(v, 0.f);   // W5[0, e] * relu(h1[e])
    }
    __syncthreads();

    if (tid == 0) {
        float s = b5[0];
        for (int k = 0; k < EDIM; ++k) s += hq[k];
        qconst[0] = s;
    }
}

// ---------------------------------------------------------------------------
// Kernel 5 (WMMA): q[j] = qconst + sum_e W5[0,E+e] * relu(h2[j,e] + b7[e]),
// with h2 = emb @ W7^T, i.e. h2[j,e] = sum_k emb[j,k] * W7[e,k].
//
// One wave32 per 16-row strip of h2 (jt = 0..63). For each of 4 column tiles
// (et), accumulate K=64 via 16x V_WMMA_F32_16X16X4_F32 (full f32 precision).
//
// f32 A 16x4 layout (ISA 7.12.2): M = lane%16, K = 2*(lane/16) + vgpr
//   -> per K-chunk kb, lane loads float2 emb[(jt*16 + lane%16)*64 + kb*4 + 2*(lane/16)]
// f32 B 4x16 layout is the transpose-symmetric striping; with
// B[k,e] = W7[et*16+e, k] the per-lane load pattern is identical on W7.
// C/D 16x16 f32: VGPR r holds (M=r, N=lane) for lanes 0-15 and (M=r+8) for
// lanes 16-31 -> row-sums are 16-lane half-wave reductions (shfl_xor 1,2,4,8).
// ---------------------------------------------------------------------------
__global__ __launch_bounds__(256) void qgemm_kernel(
    const float* __restrict__ emb,   // (N,E)
    const float* __restrict__ W7,    // (E,E)
    const float* __restrict__ b7,    // (E)
    const float* __restrict__ W5,    // (1,2E)
    const float* __restrict__ qconst_p,
    float* __restrict__ q)           // (N)
{
    const int tid  = threadIdx.x;
    const int lane = tid & 31;
    const int wave = tid >> 5;
    const int jt   = blockIdx.x * 8 + wave;   // 0..63 : 16-row strip
    const int lo   = lane & 15;
    const int hi   = lane >> 4;
    const float qc = qconst_p[0];

    const float* arow = emb + (jt * 16 + lo) * EDIM + hi * 2;

    float qacc[8];
#pragma unroll
    for (int r = 0; r < 8; ++r) qacc[r] = 0.f;

    for (int et = 0; et < 4; ++et) {
        const float* brow = W7 + (et * 16 + lo) * EDIM + hi * 2;
        v8f acc = {0.f, 0.f, 0.f, 0.f, 0.f, 0.f, 0.f, 0.f};
#pragma unroll
        for (int kb = 0; kb < 16; ++kb) {
            v2f a = *(const v2f*)(arow + kb * 4);
            v2f b = *(const v2f*)(brow + kb * 4);
            acc = __builtin_amdgcn_wmma_f32_16x16x4_f32(
                /*neg_a=*/false, a, /*neg_b=*/false, b,
                /*c_mod=*/(short)0, acc, /*reuse_a=*/false, /*reuse_b=*/false);
        }
        const int e = et * 16 + lo;
        const float bv = b7[e];
        const float wv = W5[EDIM + e];
#pragma unroll
        for (int r = 0; r < 8; ++r) {
            float t = wv * fmaxf(acc[r] + bv, 0.f);
            // reduce across the 16-lane half (rows split M=r / M=r+8)
            t += __shfl_xor(t, 1, 32);
            t += __shfl_xor(t, 2, 32);
            t += __shfl_xor(t, 4, 32);
            t += __shfl_xor(t, 8, 32);
            qacc[r] += t;
        }
    }

    if (lo == 0) {
        const int jb = jt * 16 + hi * 8;
#pragma unroll
        for (int r = 0; r < 8; ++r) q[jb + r] = qacc[r] + qc;
    }
}

// ---------------------------------------------------------------------------
extern "C" void kernel_launch(void* const* d_in, const int* in_sizes, int n_in,
                              void* d_out, int out_size, void* d_ws, size_t ws_size,
                              hipStream_t stream) {
    const float* x  = (const float*)d_in[0];   // nodes_features (1,N)
    const float* C  = (const float*)d_in[1];   // nodes_cost (N,N)
    const float* W1 = (const float*)d_in[2];
    const float* b1 = (const float*)d_in[3];
    const float* W2 = (const float*)d_in[4];
    const float* b2 = (const float*)d_in[5];
    const float* W4 = (const float*)d_in[6];
    const float* b4 = (const float*)d_in[7];
    const float* W5 = (const float*)d_in[8];
    const float* b5 = (const float*)d_in[9];
    const float* W6 = (const float*)d_in[10];
    const float* b6 = (const float*)d_in[11];
    const float* W7 = (const float*)d_in[12];
    const float* b7 = (const float*)d_in[13];
    float* out = (float*)d_out;

    float* ws     = (float*)d_ws;
    float* base   = ws;                         // N*E
    float* emb    = ws + N_NODES * EDIM;        // N*E
    float* v2     = ws + 2 * N_NODES * EDIM;    // E
    float* qconst = v2 + EDIM;                  // 1

    // base = V1 + V3
    base_kernel<<<64, 256, 0, stream>>>(x, C, W1, b1, W4, b4, base);

    // iteration 1: emb0 == 0  =>  v2 == b2
    update_kernel<<<256, 256, 0, stream>>>(base, b2, emb);

    // iterations 2..NUM_UPD
    for (int it = 1; it < NUM_UPD; ++it) {
        reduce_v2_kernel<<<1, 256, 0, stream>>>(emb, W2, b2, v2);
        update_kernel<<<256, 256, 0, stream>>>(base, v2, emb);
    }

    // qconst from the broadcast half of the head
    final_head_kernel<<<1, 256, 0, stream>>>(emb, W6, b6, W5, b5, qconst);

    // WMMA GEMM + fused relu / W5-dot -> q
    qgemm_kernel<<<8, 256, 0, stream>>>(emb, W7, b7, W5, qconst, out);
}